// TripletLoss_v4_38800734552510
// MI455X (gfx1250) — compile-verified
//
#include <hip/hip_runtime.h>

#define BN 384
#define DN 256
#define TILE 16

typedef float v2f __attribute__((ext_vector_type(2)));
typedef float v8f __attribute__((ext_vector_type(8)));

// ---------------- init: zero the global accumulators (ws is poisoned 0xAA) ----
__global__ void tl_init_acc(float* acc) {
    if (threadIdx.x < 2) acc[threadIdx.x] = 0.0f;   // acc[1] reused as uint (bits 0)
}

// ---------------- per-row L2 normalize + squared-norm of normalized row -------
__global__ __launch_bounds__(DN) void tl_normalize(const float* __restrict__ e1,
                                                   const float* __restrict__ e2,
                                                   float* __restrict__ e1n,
                                                   float* __restrict__ e2n,
                                                   float* __restrict__ a2,
                                                   float* __restrict__ b2) {
    __shared__ float red[DN];
    const int row   = blockIdx.x;
    const int which = blockIdx.y;
    const float* src = which ? e2 : e1;
    float* dst       = which ? e2n : e1n;
    float* nsqout    = which ? b2 : a2;

    float x = src[row * DN + threadIdx.x];
    red[threadIdx.x] = x * x;
    __syncthreads();
    #pragma unroll
    for (int s = DN / 2; s > 0; s >>= 1) {
        if (threadIdx.x < s) red[threadIdx.x] += red[threadIdx.x + s];
        __syncthreads();
    }
    const float nsq = red[0];
    const float inv = 1.0f / fmaxf(sqrtf(nsq), 1e-12f);
    dst[row * DN + threadIdx.x] = x * inv;
    if (threadIdx.x == 0) nsqout[row] = nsq * inv * inv;  // == ||normalized row||^2
}

// ---------------- f32 WMMA GEMM: dist tile + fused sqrt/margin epilogue -------
// One wave32 per 16x16 tile of u[i,j] = sqrt(max(a2+b2-2*dot,0)) + 0.5*ov[i,j].
// A (16x4 f32): lane m=lane&15 holds row m; VGPR0/1 = K{koff,koff+1},
// koff = (lane>=16)?2:0  -> contiguous float2 load from e1n.
// B (4x16 f32): lane holds column n=lane&15, same K split -> float2 from e2n
// (B = e2n^T, so B[k][n] = e2n[n*DN + k]).
__global__ __launch_bounds__(32) void tl_dist_wmma(const float* __restrict__ e1n,
                                                   const float* __restrict__ e2n,
                                                   const float* __restrict__ a2,
                                                   const float* __restrict__ b2,
                                                   const float* __restrict__ ov,
                                                   float* __restrict__ u) {
    const int ti   = blockIdx.x;          // 0..23
    const int tj   = blockIdx.y;          // 0..23
    const int lane = threadIdx.x;         // 0..31
    const int m    = lane & 15;
    const int koff = (lane >> 4) << 1;    // 0 or 2

    const float* arow = e1n + (ti * TILE + m) * DN + koff;
    const float* brow = e2n + (tj * TILE + m) * DN + koff;

    v8f c = {0.f, 0.f, 0.f, 0.f, 0.f, 0.f, 0.f, 0.f};
    #pragma unroll 4
    for (int k0 = 0; k0 < DN; k0 += 4) {
        v2f a = *(const v2f*)(arow + k0);
        v2f b = *(const v2f*)(brow + k0);
        // D = A(16x4,f32) x B(4x16,f32) + C  ->  v_wmma_f32_16x16x4_f32
        c = __builtin_amdgcn_wmma_f32_16x16x4_f32(false, a, false, b,
                                                  (short)0, c, false, false);
    }

    // C/D layout: VGPR r, lanes 0-15 -> M=r, lanes 16-31 -> M=r+8; N = lane&15.
    const int   gj    = tj * TILE + m;
    const float bj    = b2[gj];
    const int   mbase = ti * TILE + ((lane >> 4) << 3);
    #pragma unroll
    for (int r = 0; r < 8; ++r) {
        const int gi = mbase + r;
        float d2 = a2[gi] + bj - 2.0f * c[r];
        float d  = sqrtf(fmaxf(d2, 0.0f));
        u[gi * BN + gj] = d + 0.5f * ov[gi * BN + gj];
    }
}

// ---------------- O(B^3) masked pair reduction, LDS-cached row ----------------
// t[i,j,k] = u[i,j] - u[i,k];  mask = ov[i,j]-ov[i,k] > 0.1;
// elem = (w*|t|)^2 = (t<0 ? 0.25 : 1.0) * t^2.
__global__ __launch_bounds__(BN) void tl_triplet(const float* __restrict__ u,
                                                 const float* __restrict__ ov,
                                                 float* __restrict__ acc) {
    __shared__ float urow[BN];
    __shared__ float orow[BN];
    __shared__ float sred[BN];
    __shared__ unsigned int cred[BN];

    const int i = blockIdx.x;
    const int j = threadIdx.x;
    urow[j] = u[i * BN + j];
    orow[j] = ov[i * BN + j];
    __syncthreads();

    const float uj = urow[j];
    const float oj = orow[j];
    float        s   = 0.0f;
    unsigned int cnt = 0u;
    for (int k = 0; k < BN; ++k) {
        const float t  = uj - urow[k];
        const bool  msk = (oj - orow[k]) > 0.1f;
        const float w2 = (t < 0.0f) ? 0.25f : 1.0f;
        if (msk) { s += w2 * t * t; ++cnt; }
    }

    sred[j] = s;
    cred[j] = cnt;
    __syncthreads();
    // 384 = 256 + 128: fold tail, then power-of-two tree.
    if (j < 128) { sred[j] += sred[j + 256]; cred[j] += cred[j + 256]; }
    __syncthreads();
    #pragma unroll
    for (int st = 128; st > 0; st >>= 1) {
        if (j < st) { sred[j] += sred[j + st]; cred[j] += cred[j + st]; }
        __syncthreads();
    }
    if (j == 0) {
        atomicAdd(&acc[0], sred[0]);
        atomicAdd((unsigned int*)&acc[1], cred[0]);   // exact integer count
    }
}

// ---------------- finalize: mean over selected triplets -----------------------
__global__ void tl_finalize(const float* __restrict__ acc, float* __restrict__ out) {
    const unsigned int cnt = ((const unsigned int*)acc)[1];
    out[0] = acc[0] / (float)cnt;
}

extern "C" void kernel_launch(void* const* d_in, const int* in_sizes, int n_in,
                              void* d_out, int out_size, void* d_ws, size_t ws_size,
                              hipStream_t stream) {
    const float* e1 = (const float*)d_in[0];   // [384,256] f32
    const float* e2 = (const float*)d_in[1];   // [384,256] f32
    const float* ov = (const float*)d_in[2];   // [384,384] f32
    float* out = (float*)d_out;

    float* ws  = (float*)d_ws;
    float* e1n = ws;                  // 384*256
    float* e2n = e1n + BN * DN;       // 384*256
    float* a2  = e2n + BN * DN;       // 384
    float* b2  = a2 + BN;             // 384
    float* u   = b2 + BN;             // 384*384
    float* acc = u + BN * BN;         // 2 (f32 sum, u32 count)

    tl_init_acc<<<1, 32, 0, stream>>>(acc);
    tl_normalize<<<dim3(BN, 2), DN, 0, stream>>>(e1, e2, e1n, e2n, a2, b2);
    tl_dist_wmma<<<dim3(BN / TILE, BN / TILE), 32, 0, stream>>>(e1n, e2n, a2, b2, ov, u);
    tl_triplet<<<BN, BN, 0, stream>>>(u, ov, acc);
    tl_finalize<<<1, 1, 0, stream>>>(acc, out);
}